// HypergraphAttention_63651415326811
// MI455X (gfx1250) — compile-verified
//
#include <hip/hip_runtime.h>
#include <hip/hip_bf16.h>

// ---------------------------------------------------------------------------
// HypergraphAttention for MI455X (gfx1250, wave32)
// Z = X@P_w + b via V_WMMA_F32_16X16X4_F32 (f32 matrix path, full precision).
// s_edge folded to a matvec (Ze never materialized). Sparse softmax +
// Laplacian scatters via persistent grid-stride blocks with L2-resident f32
// atomics and global_prefetch of the next random gather row.
// ---------------------------------------------------------------------------

typedef __attribute__((ext_vector_type(2))) float v2f;
typedef __attribute__((ext_vector_type(8))) float v8f;

#define DIMF 256   // IN == OUT == 256
#define NEG_SLOPE 0.2f

// ---- ordered-uint mapping for float atomic max --------------------------
__device__ __forceinline__ unsigned flip_f32(float f) {
  unsigned u = __float_as_uint(f);
  return u ^ ((u & 0x80000000u) ? 0xFFFFFFFFu : 0x80000000u);
}
__device__ __forceinline__ float unflip_f32(unsigned u) {
  return __uint_as_float(u ^ ((u & 0x80000000u) ? 0x80000000u : 0xFFFFFFFFu));
}

// ---------------------------------------------------------------------------
// K0: q_n = P_w @ a_w[:256], q_e = P_w @ a_w[256:], c_n = P_b.a_n, c_e = P_b.a_e
// ---------------------------------------------------------------------------
__global__ __launch_bounds__(DIMF) void prep_kernel(
    const float* __restrict__ Pw, const float* __restrict__ Pb,
    const float* __restrict__ aw, float* __restrict__ qn,
    float* __restrict__ qe, float* __restrict__ cv) {
  int k = threadIdx.x;
  float sn = 0.f, se = 0.f;
  const float* prow = Pw + (size_t)k * DIMF;
  for (int j = 0; j < DIMF; ++j) {
    float p = prow[j];
    sn += p * aw[j];
    se += p * aw[DIMF + j];
  }
  qn[k] = sn;
  qe[k] = se;
  if (k == 0) {
    float cn = 0.f, ce = 0.f;
    for (int j = 0; j < DIMF; ++j) {
      cn += Pb[j] * aw[j];
      ce += Pb[j] * aw[DIMF + j];
    }
    cv[0] = cn;
    cv[1] = ce;
  }
}

// ---------------------------------------------------------------------------
// K1: Z[m,:] = X[m,:] @ W + bias  (f32 WMMA 16x16x4, wave32)
// block = 128 threads (4 waves); each block: 16 rows x 256 cols.
// wave w covers columns [64w, 64w+64) as four 16-wide n-subtiles.
// A layout: a.x=A[lo][2hi], a.y=A[lo][2hi+1]; B: b.x=W[2hi][n+lo], b.y=W[2hi+1][n+lo]
// D layout: acc[i] -> D[i+8hi][n+lo]
// ---------------------------------------------------------------------------
__global__ __launch_bounds__(128) void gemm_wmma_kernel(
    const float* __restrict__ X, const float* __restrict__ W,
    const float* __restrict__ bias, float* __restrict__ Z) {
  const int wave = threadIdx.x >> 5;
  const int lane = threadIdx.x & 31;
  const int lo = lane & 15;
  const int hi = lane >> 4;
  const size_t m0 = (size_t)blockIdx.x * 16;
  const int nb = wave * 64;

  v8f acc[4];
#pragma unroll
  for (int t = 0; t < 4; ++t) acc[t] = (v8f){};

  const float* xrow = X + (m0 + (size_t)lo) * DIMF;

  for (int k = 0; k < DIMF; k += 4) {
    const int ka = k + 2 * hi;               // A: K = 2*hi (+1 for a.y)
    v2f a = *(const v2f*)(xrow + ka);        // 8B-aligned (ka even)
    const float* wr0 = W + (size_t)ka * DIMF;      // B row K
    const float* wr1 = wr0 + DIMF;                 // B row K+1
#pragma unroll
    for (int t = 0; t < 4; ++t) {
      const int n = nb + 16 * t + lo;
      v2f b;
      b.x = wr0[n];
      b.y = wr1[n];
      acc[t] = __builtin_amdgcn_wmma_f32_16x16x4_f32(
          false, a, false, b, (short)0, acc[t], false, false);
    }
  }

  // epilogue: bias add + store
  float* zrow = Z + m0 * DIMF;
#pragma unroll
  for (int t = 0; t < 4; ++t) {
    const int n = nb + 16 * t + lo;
    const float bt = bias[n];
#pragma unroll
    for (int i = 0; i < 8; ++i) {
      zrow[(size_t)(i + 8 * hi) * DIMF + n] = acc[t][i] + bt;
    }
  }
}

// ---------------------------------------------------------------------------
// K2: out[r] = dot(A[r,:], q) + c   (one wave per row, wave32 xor-reduce)
// ---------------------------------------------------------------------------
__global__ __launch_bounds__(256) void matvec_kernel(
    const float* __restrict__ A, const float* __restrict__ q,
    const float* __restrict__ cval, float* __restrict__ out, int rows) {
  int wid = (int)((blockIdx.x * (size_t)blockDim.x + threadIdx.x) >> 5);
  int lane = threadIdx.x & 31;
  if (wid >= rows) return;
  const float* a = A + (size_t)wid * DIMF + lane * 8;
  const float* qq = q + lane * 8;
  float s = 0.f;
#pragma unroll
  for (int i = 0; i < 8; ++i) s += a[i] * qq[i];
#pragma unroll
  for (int off = 16; off; off >>= 1) s += __shfl_xor(s, off, 32);
  if (lane == 0) out[wid] = s + cval[0];
}

// ---------------------------------------------------------------------------
// K3: zero fill (graph-capture-safe init)
// ---------------------------------------------------------------------------
__global__ void zero_kernel(float* __restrict__ p, size_t n) {
  size_t i = blockIdx.x * (size_t)blockDim.x + threadIdx.x;
  size_t stride = (size_t)gridDim.x * blockDim.x;
  for (; i < n; i += stride) p[i] = 0.f;
}

// ---------------------------------------------------------------------------
// K4: sim = leaky_relu(s_node[row] + s_edge[col] + a_b); segment max per row
// ---------------------------------------------------------------------------
__global__ __launch_bounds__(256) void sim_kernel(
    const int* __restrict__ row, const int* __restrict__ col,
    const float* __restrict__ sn, const float* __restrict__ se,
    const float* __restrict__ ab, float* __restrict__ buf,
    unsigned* __restrict__ mOrd, int nnz) {
  int i = blockIdx.x * blockDim.x + threadIdx.x;
  if (i >= nnz) return;
  int r = row[i], c = col[i];
  float s = sn[r] + se[c] + ab[0];
  s = (s >= 0.f) ? s : NEG_SLOPE * s;
  buf[i] = s;
  atomicMax(&mOrd[r], flip_f32(s));
}

// K5: ex = exp(sim - m[row]); den[row] += ex
__global__ __launch_bounds__(256) void exp_kernel(
    const int* __restrict__ row, const unsigned* __restrict__ mOrd,
    float* __restrict__ buf, float* __restrict__ den, int nnz) {
  int i = blockIdx.x * blockDim.x + threadIdx.x;
  if (i >= nnz) return;
  int r = row[i];
  float e = expf(buf[i] - unflip_f32(mOrd[r]));
  buf[i] = e;
  atomicAdd(&den[r], e);
}

// K6: v = ex/den[row]; dV[row]+=v; dE[col]+=v
__global__ __launch_bounds__(256) void norm_kernel(
    const int* __restrict__ row, const int* __restrict__ col,
    const float* __restrict__ den, float* __restrict__ buf,
    float* __restrict__ dV, float* __restrict__ dE, int nnz) {
  int i = blockIdx.x * blockDim.x + threadIdx.x;
  if (i >= nnz) return;
  int r = row[i], c = col[i];
  float v = buf[i] / den[r];
  buf[i] = v;
  atomicAdd(&dV[r], v);
  atomicAdd(&dE[c], v);
}

// ---------------------------------------------------------------------------
// K7: T[col] += v * (rv[row] * Z[row])
// Persistent grid-stride blocks (256 thr = 1 feature each); prefetch the
// next iteration's randomly-gathered Z row (global_prefetch_b8, speculative).
// ---------------------------------------------------------------------------
__global__ __launch_bounds__(256) void scatter_edge_kernel(
    const int* __restrict__ row, const int* __restrict__ col,
    const float* __restrict__ vbuf, const float* __restrict__ dV,
    const float* __restrict__ Z, float* __restrict__ T, int nnz) {
  const int f = threadIdx.x;
  const int stride = gridDim.x;
  for (int i = blockIdx.x; i < nnz; i += stride) {
    if (i + stride < nnz) {
      int rn = row[i + stride];
      __builtin_prefetch(&Z[(size_t)rn * DIMF + f], 0, 1);
    }
    int r = row[i], c = col[i];
    float s = vbuf[i] * __builtin_amdgcn_rsqf(dV[r]);   // v * rv[row]
    atomicAdd(&T[(size_t)c * DIMF + f], s * Z[(size_t)r * DIMF + f]);
  }
}

// K8: T[e,:] *= 1/dE[e]   (grid-stride over E*256 elements)
__global__ __launch_bounds__(256) void scale_edge_kernel(
    float* __restrict__ T, const float* __restrict__ dE, size_t total) {
  size_t i = blockIdx.x * (size_t)blockDim.x + threadIdx.x;
  size_t stride = (size_t)gridDim.x * blockDim.x;
  for (; i < total; i += stride) T[i] *= (1.0f / dE[i >> 8]);
}

// K9: out[row] += v * T[col]   (persistent blocks + prefetch of next T row)
__global__ __launch_bounds__(256) void scatter_node_kernel(
    const int* __restrict__ row, const int* __restrict__ col,
    const float* __restrict__ vbuf, const float* __restrict__ T,
    float* __restrict__ out, int nnz) {
  const int f = threadIdx.x;
  const int stride = gridDim.x;
  for (int i = blockIdx.x; i < nnz; i += stride) {
    if (i + stride < nnz) {
      int cn = col[i + stride];
      __builtin_prefetch(&T[(size_t)cn * DIMF + f], 0, 1);
    }
    int r = row[i], c = col[i];
    atomicAdd(&out[(size_t)r * DIMF + f], vbuf[i] * T[(size_t)c * DIMF + f]);
  }
}

// K10: out[n,:] *= rv[n]   (grid-stride over N*256 elements)
__global__ __launch_bounds__(256) void scale_node_kernel(
    float* __restrict__ out, const float* __restrict__ dV, size_t total) {
  size_t i = blockIdx.x * (size_t)blockDim.x + threadIdx.x;
  size_t stride = (size_t)gridDim.x * blockDim.x;
  for (; i < total; i += stride) out[i] *= __builtin_amdgcn_rsqf(dV[i >> 8]);
}

// ---------------------------------------------------------------------------
extern "C" void kernel_launch(void* const* d_in, const int* in_sizes, int n_in,
                              void* d_out, int out_size, void* d_ws,
                              size_t ws_size, hipStream_t stream) {
  const int* row = (const int*)d_in[0];
  const int* col = (const int*)d_in[1];
  const float* X = (const float*)d_in[2];
  const float* Xe = (const float*)d_in[3];
  const float* Pw = (const float*)d_in[4];
  const float* Pb = (const float*)d_in[5];
  const float* aw = (const float*)d_in[6];
  const float* ab = (const float*)d_in[7];

  const int NNZ = in_sizes[0];
  const int N = in_sizes[2] / DIMF;
  const int E = in_sizes[3] / DIMF;

  // ---- workspace layout (floats) ----
  float* ws = (float*)d_ws;
  size_t off = 0;
  float* Z = ws + off;       off += (size_t)N * DIMF;
  float* qn = ws + off;      off += DIMF;
  float* qe = ws + off;      off += DIMF;
  float* cv = ws + off;      off += 2;
  float* s_node = ws + off;  off += N;
  float* s_edge = ws + off;  off += E;
  float* buf = ws + off;     off += NNZ;           // sim -> ex -> v (in place)
  float* zstart = ws + off;
  unsigned* mOrd = (unsigned*)(ws + off); off += N;
  float* den = ws + off;     off += N;
  float* dV = ws + off;      off += N;
  float* dE = ws + off;      off += E;
  float* T = ws + off;       off += (size_t)E * DIMF;
  size_t zcount = (size_t)((ws + off) - zstart);

  // K3: zero init (mOrd/den/dV/dE/T region and d_out accumulator)
  zero_kernel<<<2048, 256, 0, stream>>>(zstart, zcount);
  zero_kernel<<<2048, 256, 0, stream>>>((float*)d_out, (size_t)N * DIMF);

  // K0: fold a_w through P_w
  prep_kernel<<<1, DIMF, 0, stream>>>(Pw, Pb, aw, qn, qe, cv);

  // K1: Z = X@Pw + Pb  (N multiple of 16; 100000/16 = 6250)
  gemm_wmma_kernel<<<dim3(N / 16), 128, 0, stream>>>(X, Pw, Pb, Z);

  // K2: s_node = X@qn + cn ; s_edge = Xe@qe + ce
  matvec_kernel<<<dim3((N + 7) / 8), 256, 0, stream>>>(X, qn, cv, s_node, N);
  matvec_kernel<<<dim3((E + 7) / 8), 256, 0, stream>>>(Xe, qe, cv + 1, s_edge, E);

  const int nb = (NNZ + 255) / 256;
  // K4..K6: leaky-relu sim, segment max, exp-sum, normalize + degrees
  sim_kernel<<<dim3(nb), 256, 0, stream>>>(row, col, s_node, s_edge, ab, buf, mOrd, NNZ);
  exp_kernel<<<dim3(nb), 256, 0, stream>>>(row, mOrd, buf, den, NNZ);
  norm_kernel<<<dim3(nb), 256, 0, stream>>>(row, col, den, buf, dV, dE, NNZ);

  // K7..K10: Laplacian: T = (H^T D_V^-1/2 Z)_att * D_E^-1 ; out = D_V^-1/2 H T
  int sblocks = NNZ < 16384 ? NNZ : 16384;            // persistent blocks
  scatter_edge_kernel<<<dim3(sblocks), 256, 0, stream>>>(row, col, buf, dV, Z, T, NNZ);
  scale_edge_kernel<<<2048, 256, 0, stream>>>(T, dE, (size_t)E * DIMF);
  scatter_node_kernel<<<dim3(sblocks), 256, 0, stream>>>(row, col, buf, T, (float*)d_out, NNZ);
  scale_node_kernel<<<4096, 256, 0, stream>>>((float*)d_out, dV, (size_t)N * DIMF);
}